// GAT_29257317220562
// MI455X (gfx1250) — compile-verified
//
#include <hip/hip_runtime.h>
#include <hip/hip_bf16.h>

typedef __attribute__((ext_vector_type(2))) float v2f;
typedef __attribute__((ext_vector_type(8))) float v8f;

#define NN 50000
#define NE 1600000
#define FIN 128
#define NEG_SLOPE 0.2f

// ---------------------------------------------------------------- fill
__global__ void fill_f32(float* __restrict__ p, float v, int n) {
  int i = blockIdx.x * blockDim.x + threadIdx.x;
  if (i < n) p[i] = v;
}

// ---------------------------------------------------------------- GEMM (WMMA f32)
// FT[N x F_OUT] = H[N x 128] @ W[128 x F_OUT], one wave per 16-row tile,
// all F_OUT/16 column tiles accumulated in registers. EXEC is all-ones for
// every wave that reaches the WMMAs (bound check is wave-uniform).
template <int F_OUT>
__global__ __launch_bounds__(256) void gemm_wmma(const float* __restrict__ H,
                                                 const float* __restrict__ W,
                                                 float* __restrict__ FT,
                                                 int n_mtiles) {
  constexpr int NT = F_OUT / 16;
  int wave = blockIdx.x * (blockDim.x >> 5) + (threadIdx.x >> 5);
  if (wave >= n_mtiles) return;
  int lane = threadIdx.x & 31;
  int li = lane & 15;    // row (A) / col (B,C,D) within tile
  int half = lane >> 4;  // selects K pair (A/B) or row half (C/D)

  const float* __restrict__ arow = H + (size_t)(wave * 16 + li) * FIN;

  v8f acc[NT] = {};
  for (int k = 0; k < FIN; k += 4) {
    // A 16x4 f32: lane li holds row M=li; VGPR0/1 = K = 2*half, 2*half+1
    v2f a = *(const v2f*)(arow + k + 2 * half);
    const float* __restrict__ wrow = W + (size_t)(k + 2 * half) * F_OUT;
#pragma unroll
    for (int t = 0; t < NT; ++t) {
      // B 4x16 f32: lane li holds col N=li; VGPR0/1 = K = 2*half, 2*half+1
      v2f b;
      b.x = wrow[t * 16 + li];
      b.y = wrow[F_OUT + t * 16 + li];
      acc[t] = __builtin_amdgcn_wmma_f32_16x16x4_f32(
          /*neg_a=*/false, a, /*neg_b=*/false, b,
          /*c_mod=*/(short)0, acc[t], /*reuse_a=*/false, /*reuse_b=*/false);
    }
  }
  // C/D 16x16 f32: VGPR r, lane li -> row r + 8*half, col li
#pragma unroll
  for (int t = 0; t < NT; ++t)
#pragma unroll
    for (int r = 0; r < 8; ++r)
      FT[(size_t)(wave * 16 + r + 8 * half) * F_OUT + t * 16 + li] = acc[t][r];
}

// ---------------------------------------------------------------- el/er (wave per node)
template <int F>
__global__ __launch_bounds__(256) void node_el_er(const float* __restrict__ FT,
                                                  const float* __restrict__ al,
                                                  const float* __restrict__ ar,
                                                  float* __restrict__ el,
                                                  float* __restrict__ er, int n) {
  int node = blockIdx.x * (blockDim.x >> 5) + (threadIdx.x >> 5);
  if (node >= n) return;
  int lane = threadIdx.x & 31;
  float sl = 0.f, sr = 0.f;
  for (int f = lane; f < F; f += 32) {
    float v = FT[(size_t)node * F + f];
    sl += v * al[f];
    sr += v * ar[f];
  }
#pragma unroll
  for (int off = 16; off > 0; off >>= 1) {
    sl += __shfl_xor(sl, off, 32);
    sr += __shfl_xor(sr, off, 32);
  }
  if (lane == 0) { el[node] = sl; er[node] = sr; }
}

// ---------------------------------------------------------------- edge score + seg max
__device__ __forceinline__ void atomicMaxF32(float* addr, float v) {
  // sign-aware monotone reinterpretation trick; emax initialized to -inf
  if (v >= 0.f) atomicMax((int*)addr, __float_as_int(v));
  else          atomicMin((unsigned int*)addr, __float_as_uint(v));
}

__global__ __launch_bounds__(256) void edge_score_max(const int* __restrict__ src,
                                                      const int* __restrict__ dst,
                                                      const float* __restrict__ el,
                                                      const float* __restrict__ er,
                                                      float* __restrict__ ebuf,
                                                      float* __restrict__ emax, int ne) {
  int i = blockIdx.x * blockDim.x + threadIdx.x;
  if (i >= ne) return;
  int s = src[i], d = dst[i];
  float e = el[s] + er[d];
  e = (e > 0.f) ? e : NEG_SLOPE * e;  // LeakyReLU
  ebuf[i] = e;
  atomicMaxF32(emax + d, e);
}

// ---------------------------------------------------------------- exp + seg sum
__global__ __launch_bounds__(256) void edge_exp_sum(const int* __restrict__ dst,
                                                    const float* __restrict__ emax,
                                                    float* __restrict__ ebuf,
                                                    float* __restrict__ esum, int ne) {
  int i = blockIdx.x * blockDim.x + threadIdx.x;
  if (i >= ne) return;
  int d = dst[i];
  float w = __expf(ebuf[i] - emax[d]);
  ebuf[i] = w;
  atomicAdd(esum + d, w);
}

// ---------------------------------------------------------------- weighted scatter-add
// F/4 threads per edge, each handles a float4 chunk of ft[src]; for F=128 a
// full wave covers one 512B row -> coalesced gather, f32 atomic scatter.
template <int F>
__global__ __launch_bounds__(256) void edge_aggregate(const int* __restrict__ src,
                                                      const int* __restrict__ dst,
                                                      const float* __restrict__ ebuf,
                                                      const float* __restrict__ ft,
                                                      float* __restrict__ agg, int ne) {
  constexpr int CH = F / 4;
  int t = blockIdx.x * blockDim.x + threadIdx.x;
  int edge = t / CH;
  int c = t - edge * CH;
  if (edge >= ne) return;
  int s = src[edge], d = dst[edge];
  float w = ebuf[edge];
  float4 v = *(const float4*)(ft + (size_t)s * F + c * 4);
  float* o = agg + (size_t)d * F + c * 4;
  atomicAdd(o + 0, w * v.x);
  atomicAdd(o + 1, w * v.y);
  atomicAdd(o + 2, w * v.z);
  atomicAdd(o + 3, w * v.w);
}

// ---------------------------------------------------------------- normalize + bias + relu
template <int F>
__global__ __launch_bounds__(256) void finalize_k(const float* __restrict__ agg,
                                                  const float* __restrict__ esum,
                                                  const float* __restrict__ b,
                                                  float* __restrict__ out, int n) {
  int t = blockIdx.x * blockDim.x + threadIdx.x;
  if (t >= n * F) return;
  int node = t / F;
  int f = t - node * F;
  float s = esum[node];
  s = (s > 0.f) ? s : 1.f;
  float v = agg[t] / s + b[f];
  out[t] = (v > 0.f) ? v : 0.f;
}

// ---------------------------------------------------------------- one GAT layer
template <int F_OUT>
static void gat_layer(const float* h_in, const int* src, const int* dst,
                      const float* W, const float* al, const float* ar, const float* b,
                      float* h_out, float* ft, float* agg, float* el, float* er,
                      float* emax, float* esum, float* ebuf, hipStream_t stream) {
  const int MT = NN / 16;  // 3125, exact
  gemm_wmma<F_OUT><<<(MT + 7) / 8, 256, 0, stream>>>(h_in, W, ft, MT);
  node_el_er<F_OUT><<<(NN + 7) / 8, 256, 0, stream>>>(ft, al, ar, el, er, NN);
  fill_f32<<<(NN + 255) / 256, 256, 0, stream>>>(emax, -__builtin_huge_valf(), NN);
  fill_f32<<<(NN + 255) / 256, 256, 0, stream>>>(esum, 0.f, NN);
  fill_f32<<<(NN * F_OUT + 255) / 256, 256, 0, stream>>>(agg, 0.f, NN * F_OUT);
  edge_score_max<<<(NE + 255) / 256, 256, 0, stream>>>(src, dst, el, er, ebuf, emax, NE);
  edge_exp_sum<<<(NE + 255) / 256, 256, 0, stream>>>(dst, emax, ebuf, esum, NE);
  long total = (long)NE * (F_OUT / 4);
  edge_aggregate<F_OUT><<<(int)((total + 255) / 256), 256, 0, stream>>>(src, dst, ebuf, ft, agg, NE);
  finalize_k<F_OUT><<<(NN * F_OUT + 255) / 256, 256, 0, stream>>>(agg, esum, b, h_out, NN);
}

extern "C" void kernel_launch(void* const* d_in, const int* in_sizes, int n_in,
                              void* d_out, int out_size, void* d_ws, size_t ws_size,
                              hipStream_t stream) {
  const float* feat = (const float*)d_in[0];
  const int* src = (const int*)d_in[1];
  const int* dst = (const int*)d_in[2];
  const float* W1 = (const float*)d_in[3];
  const float* al1 = (const float*)d_in[4];
  const float* ar1 = (const float*)d_in[5];
  const float* b1 = (const float*)d_in[6];
  const float* W2 = (const float*)d_in[7];
  const float* al2 = (const float*)d_in[8];
  const float* ar2 = (const float*)d_in[9];
  const float* b2 = (const float*)d_in[10];
  const float* W3 = (const float*)d_in[11];
  const float* al3 = (const float*)d_in[12];
  const float* ar3 = (const float*)d_in[13];
  const float* b3 = (const float*)d_in[14];

  float* ws = (float*)d_ws;
  size_t off = 0;
  float* ft   = ws + off; off += (size_t)NN * 128;
  float* agg  = ws + off; off += (size_t)NN * 128;
  float* hbuf = ws + off; off += (size_t)NN * 128;  // h1 then h2 (gemm consumes before finalize writes)
  float* ebuf = ws + off; off += (size_t)NE;
  float* el   = ws + off; off += NN;
  float* er   = ws + off; off += NN;
  float* emax = ws + off; off += NN;
  float* esum = ws + off; off += NN;

  gat_layer<128>(feat, src, dst, W1, al1, ar1, b1, hbuf, ft, agg, el, er, emax, esum, ebuf, stream);
  gat_layer<128>(hbuf, src, dst, W2, al2, ar2, b2, hbuf, ft, agg, el, er, emax, esum, ebuf, stream);
  gat_layer<16>(hbuf, src, dst, W3, al3, ar3, b3, (float*)d_out, ft, agg, el, er, emax, esum, ebuf, stream);
}